// PointNetChebDconv_61546881352033
// MI455X (gfx1250) — compile-verified
//
#include <hip/hip_runtime.h>

// ---------------------------------------------------------------- types
typedef float v2f __attribute__((ext_vector_type(2)));
typedef float v8f __attribute__((ext_vector_type(8)));

__device__ __forceinline__ v8f wmma_f32_4(v2f a, v2f b, v8f c) {
  // D = A(16x4) * B(4x16) + C(16x16), fp32 WMMA
  return __builtin_amdgcn_wmma_f32_16x16x4_f32(false, a, false, b, (short)0, c,
                                               false, false);
}

__device__ __forceinline__ float dist2_nc(float ax, float ay, float az,
                                          float bx, float by, float bz) {
  float dx = ax - bx, dy = ay - by, dz = az - bz;
  return __fadd_rn(__fadd_rn(__fmul_rn(dx, dx), __fmul_rn(dy, dy)),
                   __fmul_rn(dz, dz));
}

// ---------------------------------------------------------------- generic WMMA GEMM
// Y[M,N] = act(X[M,K(ldx)] @ W[K,N] + bias)
// mode 0: +bias                      mode 1: relu((.+b)*g+be)
// mode 2: (relu((.+b)*g+be))*g2+be2  mode 4: relu(.+b)
// accum: init accumulator from Y (Y += X@W ...)
__global__ void __launch_bounds__(128)
gemm_kernel(const float* __restrict__ X, const float* __restrict__ W,
            const float* __restrict__ bias, const float* __restrict__ gam,
            const float* __restrict__ bet, const float* __restrict__ gam2,
            const float* __restrict__ bet2, float* __restrict__ Y,
            int M, int N, int K, int Kreal, int ldx, int mode, int accum) {
  int ntN = (N + 15) >> 4;
  int ntM = (M + 15) >> 4;
  int tile = blockIdx.x * 4 + (threadIdx.x >> 5);  // one 16x16 tile per wave
  if (tile >= ntM * ntN) return;                   // wave-uniform exit
  int mt = tile / ntN, nt = tile % ntN;
  int m0 = mt << 4, n0 = nt << 4;
  int lane = threadIdx.x & 31;
  int lm = lane & 15, hg = lane >> 4;

  int row = m0 + lm; if (row >= M) row = M - 1;    // clamp (results discarded)
  int col = n0 + lm;
  bool cok = col < N;
  const float* Xr = X + (size_t)row * ldx;

  v8f acc = {0.f, 0.f, 0.f, 0.f, 0.f, 0.f, 0.f, 0.f};
  if (accum) {
    int colc0 = cok ? col : N - 1;
    for (int v = 0; v < 8; ++v) {
      int rr = m0 + v + (hg << 3);
      int rrc = rr < M ? rr : M - 1;
      float val = Y[(size_t)rrc * N + colc0];
      acc[v] = (rr < M && cok) ? val : 0.f;
    }
  }

  if (Kreal == K && n0 + 16 <= N) {
    // ---- fast path: no bounds handling inside the K loop ----
    const float* Wc = W + col;
    #pragma unroll 4
    for (int k0 = 0; k0 < K; k0 += 4) {
      int ka = k0 + (hg << 1);               // A frag: lane-half holds K=2g,2g+1
      v2f a = *(const v2f*)(Xr + ka);        // contiguous pair -> b64 load
      v2f b;
      b.x = Wc[(size_t)ka * N];
      b.y = Wc[(size_t)(ka + 1) * N];
      acc = wmma_f32_4(a, b, acc);
    }
  } else {
    // ---- guarded path: clamp addresses, mask values (no exec divergence) ----
    int colc = cok ? col : N - 1;
    const float* Wc = W + colc;
    for (int k0 = 0; k0 < K; k0 += 4) {
      int ka = k0 + (hg << 1);
      v2f a = *(const v2f*)(Xr + ka);
      int ka0 = ka < Kreal ? ka : Kreal - 1;
      int ka1 = ka + 1 < Kreal ? ka + 1 : Kreal - 1;
      float w0 = Wc[(size_t)ka0 * N];
      float w1 = Wc[(size_t)ka1 * N];
      v2f b;
      b.x = (cok && ka < Kreal) ? w0 : 0.f;
      b.y = (cok && ka + 1 < Kreal) ? w1 : 0.f;
      acc = wmma_f32_4(a, b, acc);
    }
  }

  float bb = (bias && cok) ? bias[col] : 0.f;
  float ga = 1.f, be_ = 0.f, g2 = 1.f, b2 = 0.f;
  if ((mode == 1 || mode == 2) && cok) { ga = gam[col]; be_ = bet[col]; }
  if (mode == 2 && cok) { g2 = gam2[col]; b2 = bet2[col]; }

  for (int v = 0; v < 8; ++v) {
    int rr = m0 + v + (hg << 3);
    if (rr < M && cok) {
      float y = acc[v] + bb;
      if (mode == 1 || mode == 2) y = fmaxf(y * ga + be_, 0.f);
      if (mode == 2) y = y * g2 + b2;
      if (mode == 4) y = fmaxf(y, 0.f);
      Y[(size_t)rr * N + col] = y;
    }
  }
}

// ---------------------------------------------------------------- deform net (3->32->64->3, tanh*0.1 residual)
__global__ void deform_kernel(const float* __restrict__ pos, float* __restrict__ dpos, int npts,
                              const float* W0, const float* b0, const float* be0, const float* g0,
                              const float* W1, const float* b1, const float* be1, const float* g1,
                              const float* W2, const float* b2, const float* be2, const float* g2) {
  int i = blockIdx.x * blockDim.x + threadIdx.x;
  if (i >= npts) return;
  float in0 = pos[3 * i], in1 = pos[3 * i + 1], in2 = pos[3 * i + 2];
  float h1[32];
  #pragma unroll
  for (int o = 0; o < 32; ++o) {
    float s = b0[o] + in0 * W0[o] + in1 * W0[32 + o] + in2 * W0[64 + o];
    h1[o] = fmaxf(s * g0[o] + be0[o], 0.f);
  }
  float h2[64];
  for (int o = 0; o < 64; ++o) {
    float s = b1[o];
    #pragma unroll
    for (int k = 0; k < 32; ++k) s += h1[k] * W1[k * 64 + o];
    h2[o] = fmaxf(s * g1[o] + be1[o], 0.f);
  }
  for (int o = 0; o < 3; ++o) {
    float s = b2[o];
    #pragma unroll
    for (int k = 0; k < 64; ++k) s += h2[k] * W2[k * 3 + o];
    dpos[3 * i + o] = pos[3 * i + o] + tanhf(s) * 0.1f;
  }
}

// ---------------------------------------------------------------- farthest point sampling (1 block / batch)
#define FPS_MAXN 2048
__global__ void fps_kernel(const float* __restrict__ pos, int n, int m,
                           int* __restrict__ idx_out, float* __restrict__ posq) {
  int b = blockIdx.x, t = threadIdx.x;
  const float* P = pos + (size_t)b * n * 3;
  __shared__ float px[FPS_MAXN], py[FPS_MAXN], pz[FPS_MAXN], dd[FPS_MAXN];
  __shared__ float rv[256];
  __shared__ int ri[256];
  for (int i = t; i < n; i += 256) {
    px[i] = P[3 * i]; py[i] = P[3 * i + 1]; pz[i] = P[3 * i + 2];
  }
  __syncthreads();
  float cx = px[0], cy = py[0], cz = pz[0];
  for (int i = t; i < n; i += 256) dd[i] = dist2_nc(px[i], py[i], pz[i], cx, cy, cz);
  if (t == 0) idx_out[(size_t)b * m] = 0;
  __syncthreads();
  for (int it = 1; it < m; ++it) {
    float bv = -1.f; int bi = 0;
    for (int i = t; i < n; i += 256) {               // ascending scan: first-max kept
      float v = dd[i];
      if (v > bv) { bv = v; bi = i; }
    }
    rv[t] = bv; ri[t] = bi;
    __syncthreads();
    for (int s = 128; s > 0; s >>= 1) {
      if (t < s) {
        float v2 = rv[t + s]; int i2 = ri[t + s];
        if (v2 > rv[t] || (v2 == rv[t] && i2 < ri[t])) { rv[t] = v2; ri[t] = i2; }
      }
      __syncthreads();
    }
    int cur = ri[0];
    if (t == 0) idx_out[(size_t)b * m + it] = cur;
    float nx = px[cur], ny = py[cur], nz = pz[cur];
    for (int i = t; i < n; i += 256) {
      float v = dist2_nc(px[i], py[i], pz[i], nx, ny, nz);
      if (v < dd[i]) dd[i] = v;
    }
    __syncthreads();
  }
  const int* I = idx_out + (size_t)b * m;
  for (int q = t; q < m; q += 256) {
    int j = I[q];
    float* o = posq + ((size_t)b * m + q) * 3;
    o[0] = px[j]; o[1] = py[j]; o[2] = pz[j];
  }
}

// ---------------------------------------------------------------- k-smallest neighbors (1 block / query)
__global__ void topk_kernel(const float* __restrict__ pts, const float* __restrict__ qry,
                            int n, int mq, int kk,
                            int* __restrict__ nbr, float* __restrict__ nbrd) {
  int gq = blockIdx.x, t = threadIdx.x;
  int b = gq / mq;
  const float* P = pts + (size_t)b * n * 3;
  const float* Q = qry + (size_t)gq * 3;
  __shared__ float dd[FPS_MAXN];
  __shared__ float rv[256];
  __shared__ int ri[256];
  float qx = Q[0], qy = Q[1], qz = Q[2];
  for (int i = t; i < n; i += 256)
    dd[i] = dist2_nc(P[3 * i], P[3 * i + 1], P[3 * i + 2], qx, qy, qz);
  __syncthreads();
  for (int it = 0; it < kk; ++it) {
    float bv = 1e30f; int bi = 0;
    for (int i = t; i < n; i += 256) {               // ascending: first-min kept
      float v = dd[i];
      if (v < bv) { bv = v; bi = i; }
    }
    rv[t] = bv; ri[t] = bi;
    __syncthreads();
    for (int s = 128; s > 0; s >>= 1) {
      if (t < s) {
        float v2 = rv[t + s]; int i2 = ri[t + s];
        if (v2 < rv[t] || (v2 == rv[t] && i2 < ri[t])) { rv[t] = v2; ri[t] = i2; }
      }
      __syncthreads();
    }
    if (t == 0) {
      nbr[(size_t)gq * kk + it] = ri[0];
      nbrd[(size_t)gq * kk + it] = rv[0];
      dd[ri[0]] = 1e30f;
    }
    __syncthreads();
  }
}

// ---------------------------------------------------------------- SA feature build: [x(nbr), dpos(nbr)-posq] zero-padded
__global__ void safeats_kernel(const float* __restrict__ X, int Cx,
                               const float* __restrict__ dpos, const float* __restrict__ posq,
                               const int* __restrict__ nbr, int q0, int nq, int mq, int np,
                               int Kpad, float* __restrict__ feats) {
  int i = blockIdx.x * blockDim.x + threadIdx.x;
  int total = nq * 64 * Kpad;
  if (i >= total) return;
  int c = i % Kpad, r = i / Kpad;
  int j = r & 63, ql = r >> 6;
  int g = q0 + ql, b = g / mq;
  int p = nbr[(size_t)g * 64 + j];
  float v = 0.f;
  if (c < Cx) v = X[((size_t)b * np + p) * Cx + c];
  else if (c < Cx + 3) {
    int d = c - Cx;
    v = dpos[((size_t)b * np + p) * 3 + d] - posq[(size_t)g * 3 + d];
  }
  feats[(size_t)r * Kpad + c] = v;
}

// ---------------------------------------------------------------- SA masked max over 64 neighbors
__global__ void samax_kernel(const float* __restrict__ h3, const float* __restrict__ nbrd,
                             int q0, int nq, int C, float r2, float* __restrict__ out) {
  int i = blockIdx.x * blockDim.x + threadIdx.x;
  if (i >= nq * C) return;
  int ql = i / C, c = i % C;
  int g = q0 + ql;
  const float* dp = nbrd + (size_t)g * 64;
  const float* hp = h3 + ((size_t)ql * 64) * C + c;
  float mv = -1e30f; int any = 0;
  for (int j = 0; j < 64; ++j) {
    if (dp[j] <= r2) { any = 1; mv = fmaxf(mv, hp[(size_t)j * C]); }
  }
  out[(size_t)g * C + c] = any ? mv : 0.f;
}

// ---------------------------------------------------------------- transition-down gather + max over k
__global__ void gathermax_kernel(const float* __restrict__ H, const int* __restrict__ nbr,
                                 int mq, int np, int k, int C, float* __restrict__ out, int total) {
  int i = blockIdx.x * blockDim.x + threadIdx.x;
  if (i >= total) return;
  int g = i / C, c = i % C;
  int b = g / mq;
  const int* nb = nbr + (size_t)g * k;
  float mv = -3.4e38f;
  for (int j = 0; j < k; ++j) {
    int p = nb[j];
    mv = fmaxf(mv, H[((size_t)b * np + p) * C + c]);
  }
  out[(size_t)g * C + c] = mv;
}

// ---------------------------------------------------------------- Cheb scatter:  out[nbr[g][j]] += x[g]*scale
__global__ void chebscat_kernel(const float* __restrict__ x, const int* __restrict__ nbr,
                                int mq, int k, int C, float scale, float* __restrict__ out, int total) {
  int i = blockIdx.x * blockDim.x + threadIdx.x;
  if (i >= total) return;
  int c = i % C;
  int rest = i / C;
  int j = rest % k;
  int g = rest / k;
  int b = g / mq;
  int dst = nbr[(size_t)g * k + j];
  atomicAdd(&out[((size_t)b * mq + dst) * C + c], x[(size_t)g * C + c] * scale);
}

// ---------------------------------------------------------------- elementwise helpers
__global__ void fill_kernel(float* p, float v, int n) {
  int i = blockIdx.x * blockDim.x + threadIdx.x;
  if (i < n) p[i] = v;
}
__global__ void t2comb_kernel(const float* tmp, const float* x, float* out, int n) {
  int i = blockIdx.x * blockDim.x + threadIdx.x;
  if (i < n) out[i] = 2.f * tmp[i] - x[i];
}
__global__ void resbn_kernel(float* x, const float* y, const float* g, const float* be, int n, int C) {
  int i = blockIdx.x * blockDim.x + threadIdx.x;
  if (i < n) { int c = i % C; x[i] = x[i] + (y[i] * g[c] + be[c]); }
}
__global__ void concat3_kernel(const float* x, const float* pos, float* out, int rows, int Cx) {
  int Kp = Cx + 4;
  int i = blockIdx.x * blockDim.x + threadIdx.x;
  if (i >= rows * Kp) return;
  int r = i / Kp, c = i % Kp;
  float v = 0.f;
  if (c < Cx) v = x[(size_t)r * Cx + c];
  else if (c < Cx + 3) v = pos[(size_t)r * 3 + (c - Cx)];
  out[i] = v;
}
__global__ void mean_kernel(const float* h, float* out, int Bn, int T, int C) {
  int i = blockIdx.x * blockDim.x + threadIdx.x;
  if (i >= Bn * C) return;
  int b = i / C, c = i % C;
  float s = 0.f;
  for (int t = 0; t < T; ++t) s += h[((size_t)b * T + t) * C + c];
  out[i] = s * (1.f / T);
}

// ---------------------------------------------------------------- host orchestration
static inline int cdiv(int a, int b) { return (a + b - 1) / b; }

extern "C" void kernel_launch(void* const* d_in, const int* in_sizes, int n_in,
                              void* d_out, int out_size, void* d_ws, size_t ws_size,
                              hipStream_t stream) {
  (void)in_sizes; (void)n_in; (void)out_size; (void)ws_size;
  auto F = [&](int i) -> const float* { return (const float*)d_in[i]; };

  // pytree-flatten (sorted dict keys) input indices
  enum {
    IN_DATA = 0, IN_BN1_BE = 1, IN_BN1_G = 2, IN_BN2_BE = 3, IN_BN2_G = 4,
    IN_CHEB1_W = 5, IN_CHEB1_B = 6, IN_CHEB2_W = 7, IN_CHEB2_B = 8,
    IN_D1 = 9, IN_D2 = 21, IN_D3 = 33, IN_D4 = 45,
    IN_HEAD = 57, IN_SA1 = 69, IN_SA2 = 81, IN_SA3 = 93,
    IN_TD1BN_BE = 105, IN_TD1BN_G = 106, IN_TD1 = 107,
    IN_TD2BN_BE = 111, IN_TD2BN_G = 112, IN_TD2 = 113
  };
  auto LW  = [&](int base, int l) { return F(base + 4 * l + 0); };
  auto LB  = [&](int base, int l) { return F(base + 4 * l + 1); };
  auto LBE = [&](int base, int l) { return F(base + 4 * l + 2); };
  auto LG  = [&](int base, int l) { return F(base + 4 * l + 3); };

  const int Bb = 8, N1 = 2048, M1 = 1024, M2 = 256, M3 = 64, M4 = 16, CQ = 256;
  const float* data = F(IN_DATA);

  // ---- workspace carve-up (fp32 / int32) ----
  char* wp = (char*)d_ws;
  auto AFm = [&](size_t n) -> float* { float* p = (float*)wp; wp += ((n * 4 + 255) / 256) * 256; return p; };
  auto AIm = [&](size_t n) -> int*   { int*   p = (int*)wp;   wp += ((n * 4 + 255) / 256) * 256; return p; };

  float* dpos1 = AFm((size_t)Bb * N1 * 3);
  int*   idx1  = AIm((size_t)Bb * M1);
  float* posq1 = AFm((size_t)Bb * M1 * 3);
  int*   nbr1  = AIm((size_t)Bb * M1 * 64);
  float* nbrd1 = AFm((size_t)Bb * M1 * 64);
  float* xsa1  = AFm((size_t)Bb * M1 * 128);
  float* dpos2 = AFm((size_t)Bb * M1 * 3);
  int*   idx2  = AIm((size_t)Bb * M2);
  float* posq2 = AFm((size_t)Bb * M2 * 3);
  int*   nbr2  = AIm((size_t)Bb * M2 * 16);
  float* htd1  = AFm((size_t)Bb * M1 * 256);
  float* xtd1  = AFm((size_t)Bb * M2 * 256);
  int*   cnbr1 = AIm((size_t)Bb * M2 * 16);
  float* t1a   = AFm((size_t)Bb * M2 * 256);
  float* t2a   = AFm((size_t)Bb * M2 * 256);
  float* tmpa  = AFm((size_t)Bb * M2 * 256);
  float* ya    = AFm((size_t)Bb * M2 * 256);
  float* dpos3 = AFm((size_t)Bb * M2 * 3);
  int*   idx3  = AIm((size_t)Bb * M3);
  float* posq3 = AFm((size_t)Bb * M3 * 3);
  int*   nbr3  = AIm((size_t)Bb * M3 * 64);
  float* nbrd3 = AFm((size_t)Bb * M3 * 64);
  float* xsa2  = AFm((size_t)Bb * M3 * 256);
  float* dpos4 = AFm((size_t)Bb * M3 * 3);
  int*   idx4  = AIm((size_t)Bb * M4);
  float* posq4 = AFm((size_t)Bb * M4 * 3);
  int*   nbr4  = AIm((size_t)Bb * M4 * 16);
  float* htd2  = AFm((size_t)Bb * M3 * 512);
  float* xtd2  = AFm((size_t)Bb * M4 * 512);
  int*   cnbr2 = AIm((size_t)Bb * M4 * 16);
  float* t1b   = AFm((size_t)Bb * M4 * 512);
  float* t2b   = AFm((size_t)Bb * M4 * 512);
  float* tmpb  = AFm((size_t)Bb * M4 * 512);
  float* yb    = AFm((size_t)Bb * M4 * 512);
  float* f3    = AFm((size_t)Bb * M4 * 516);
  float* h3a   = AFm((size_t)Bb * M4 * 256);
  float* h3b   = AFm((size_t)Bb * M4 * 512);
  float* h3c   = AFm((size_t)Bb * M4 * 1024);
  float* pooled= AFm((size_t)Bb * 1024);
  float* hh1   = AFm((size_t)Bb * 512);
  float* hh2   = AFm((size_t)Bb * 256);
  float* nbrdk = AFm((size_t)Bb * M2 * 16);      // scratch d2 for kNN calls
  float* scrF  = AFm((size_t)CQ * 64 * 260);
  float* scrH1 = AFm((size_t)CQ * 64 * 128);
  float* scrH2 = AFm((size_t)CQ * 64 * 128);
  float* scrH3 = AFm((size_t)CQ * 64 * 256);

  // ---- launch helpers ----
  auto gemm = [&](const float* X, const float* W, const float* bias, const float* g1,
                  const float* be1, const float* g2, const float* be2, float* Y,
                  int M, int N, int K, int Kreal, int ldx, int mode, int accum) {
    int tiles = cdiv(M, 16) * cdiv(N, 16);
    gemm_kernel<<<cdiv(tiles, 4), 128, 0, stream>>>(X, W, bias, g1, be1, g2, be2, Y,
                                                    M, N, K, Kreal, ldx, mode, accum);
  };
  auto deform = [&](const float* pos, float* dpos, int npts, int base) {
    deform_kernel<<<cdiv(npts, 256), 256, 0, stream>>>(pos, dpos, npts,
        LW(base, 0), LB(base, 0), LBE(base, 0), LG(base, 0),
        LW(base, 1), LB(base, 1), LBE(base, 1), LG(base, 1),
        LW(base, 2), LB(base, 2), LBE(base, 2), LG(base, 2));
  };
  auto fill = [&](float* p, int n) { fill_kernel<<<cdiv(n, 256), 256, 0, stream>>>(p, 0.f, n); };

  // ======================= SA module 1 (x = data, pos = data) =======================
  deform(data, dpos1, Bb * N1, IN_D1);
  fps_kernel<<<Bb, 256, 0, stream>>>(dpos1, N1, M1, idx1, posq1);
  topk_kernel<<<Bb * M1, 256, 0, stream>>>(dpos1, posq1, N1, M1, 64, nbr1, nbrd1);
  for (int q0 = 0; q0 < Bb * M1; q0 += CQ) {
    int rows = CQ * 64;
    safeats_kernel<<<cdiv(CQ * 64 * 8, 256), 256, 0, stream>>>(data, 3, dpos1, posq1,
        nbr1, q0, CQ, M1, N1, 8, scrF);
    gemm(scrF,  LW(IN_SA1, 0), LB(IN_SA1, 0), LG(IN_SA1, 0), LBE(IN_SA1, 0), 0, 0,
         scrH1, rows, 64, 8, 6, 8, 1, 0);
    gemm(scrH1, LW(IN_SA1, 1), LB(IN_SA1, 1), LG(IN_SA1, 1), LBE(IN_SA1, 1), 0, 0,
         scrH2, rows, 64, 64, 64, 64, 1, 0);
    gemm(scrH2, LW(IN_SA1, 2), LB(IN_SA1, 2), 0, 0, 0, 0,
         scrH3, rows, 128, 64, 64, 64, 0, 0);
    samax_kernel<<<cdiv(CQ * 128, 256), 256, 0, stream>>>(scrH3, nbrd1, q0, CQ, 128,
        0.2f * 0.2f, xsa1);
  }

  // ======================= transition down 1 =======================
  deform(posq1, dpos2, Bb * M1, IN_D2);
  fps_kernel<<<Bb, 256, 0, stream>>>(dpos2, M1, M2, idx2, posq2);
  topk_kernel<<<Bb * M2, 256, 0, stream>>>(dpos2, posq2, M1, M2, 16, nbr2, nbrdk);
  gemm(xsa1, LW(IN_TD1, 0), LB(IN_TD1, 0), LG(IN_TD1, 0), LBE(IN_TD1, 0),
       F(IN_TD1BN_G), F(IN_TD1BN_BE), htd1, Bb * M1, 256, 128, 128, 128, 2, 0);
  gathermax_kernel<<<cdiv(Bb * M2 * 256, 256), 256, 0, stream>>>(htd1, nbr2, M2, M1, 16,
      256, xtd1, Bb * M2 * 256);

  // ======================= ChebConv 1 + residual BN =======================
  topk_kernel<<<Bb * M2, 256, 0, stream>>>(posq2, posq2, M2, M2, 16, cnbr1, nbrdk);
  {
    int nel = Bb * M2 * 256;
    fill(t1a, nel);
    chebscat_kernel<<<cdiv(Bb * M2 * 16 * 256, 256), 256, 0, stream>>>(xtd1, cnbr1, M2, 16,
        256, -1.f / 16.f, t1a, Bb * M2 * 16 * 256);
    fill(tmpa, nel);
    chebscat_kernel<<<cdiv(Bb * M2 * 16 * 256, 256), 256, 0, stream>>>(t1a, cnbr1, M2, 16,
        256, -1.f / 16.f, tmpa, Bb * M2 * 16 * 256);
    t2comb_kernel<<<cdiv(nel, 256), 256, 0, stream>>>(tmpa, xtd1, t2a, nel);
    const float* cw = F(IN_CHEB1_W);
    gemm(xtd1, cw,           F(IN_CHEB1_B), 0, 0, 0, 0, ya, Bb * M2, 256, 256, 256, 256, 0, 0);
    gemm(t1a,  cw + 65536,   0,             0, 0, 0, 0, ya, Bb * M2, 256, 256, 256, 256, 0, 1);
    gemm(t2a,  cw + 131072,  0,             0, 0, 0, 0, ya, Bb * M2, 256, 256, 256, 256, 0, 1);
    resbn_kernel<<<cdiv(nel, 256), 256, 0, stream>>>(xtd1, ya, F(IN_BN1_G), F(IN_BN1_BE),
        nel, 256);
  }

  // ======================= SA module 2 (x = xtd1, pos = posq2) =======================
  deform(posq2, dpos3, Bb * M2, IN_D3);
  fps_kernel<<<Bb, 256, 0, stream>>>(dpos3, M2, M3, idx3, posq3);
  topk_kernel<<<Bb * M3, 256, 0, stream>>>(dpos3, posq3, M2, M3, 64, nbr3, nbrd3);
  for (int q0 = 0; q0 < Bb * M3; q0 += CQ) {
    int rows = CQ * 64;
    safeats_kernel<<<cdiv(CQ * 64 * 260, 256), 256, 0, stream>>>(xtd1, 256, dpos3, posq3,
        nbr3, q0, CQ, M3, M2, 260, scrF);
    gemm(scrF,  LW(IN_SA2, 0), LB(IN_SA2, 0), LG(IN_SA2, 0), LBE(IN_SA2, 0), 0, 0,
         scrH1, rows, 128, 260, 259, 260, 1, 0);
    gemm(scrH1, LW(IN_SA2, 1), LB(IN_SA2, 1), LG(IN_SA2, 1), LBE(IN_SA2, 1), 0, 0,
         scrH2, rows, 128, 128, 128, 128, 1, 0);
    gemm(scrH2, LW(IN_SA2, 2), LB(IN_SA2, 2), 0, 0, 0, 0,
         scrH3, rows, 256, 128, 128, 128, 0, 0);
    samax_kernel<<<cdiv(CQ * 256, 256), 256, 0, stream>>>(scrH3, nbrd3, q0, CQ, 256,
        0.4f * 0.4f, xsa2);
  }

  // ======================= transition down 2 =======================
  deform(posq3, dpos4, Bb * M3, IN_D4);
  fps_kernel<<<Bb, 256, 0, stream>>>(dpos4, M3, M4, idx4, posq4);
  topk_kernel<<<Bb * M4, 256, 0, stream>>>(dpos4, posq4, M3, M4, 16, nbr4, nbrdk);
  gemm(xsa2, LW(IN_TD2, 0), LB(IN_TD2, 0), LG(IN_TD2, 0), LBE(IN_TD2, 0),
       F(IN_TD2BN_G), F(IN_TD2BN_BE), htd2, Bb * M3, 512, 256, 256, 256, 2, 0);
  gathermax_kernel<<<cdiv(Bb * M4 * 512, 256), 256, 0, stream>>>(htd2, nbr4, M4, M3, 16,
      512, xtd2, Bb * M4 * 512);

  // ======================= ChebConv 2 + residual BN =======================
  topk_kernel<<<Bb * M4, 256, 0, stream>>>(posq4, posq4, M4, M4, 16, cnbr2, nbrdk);
  {
    int nel = Bb * M4 * 512;
    fill(t1b, nel);
    chebscat_kernel<<<cdiv(Bb * M4 * 16 * 512, 256), 256, 0, stream>>>(xtd2, cnbr2, M4, 16,
        512, -1.f / 16.f, t1b, Bb * M4 * 16 * 512);
    fill(tmpb, nel);
    chebscat_kernel<<<cdiv(Bb * M4 * 16 * 512, 256), 256, 0, stream>>>(t1b, cnbr2, M4, 16,
        512, -1.f / 16.f, tmpb, Bb * M4 * 16 * 512);
    t2comb_kernel<<<cdiv(nel, 256), 256, 0, stream>>>(tmpb, xtd2, t2b, nel);
    const float* cw = F(IN_CHEB2_W);
    gemm(xtd2, cw,          F(IN_CHEB2_B), 0, 0, 0, 0, yb, Bb * M4, 512, 512, 512, 512, 0, 0);
    gemm(t1b,  cw + 262144, 0,             0, 0, 0, 0, yb, Bb * M4, 512, 512, 512, 512, 0, 1);
    gemm(t2b,  cw + 524288, 0,             0, 0, 0, 0, yb, Bb * M4, 512, 512, 512, 512, 0, 1);
    resbn_kernel<<<cdiv(nel, 256), 256, 0, stream>>>(xtd2, yb, F(IN_BN2_G), F(IN_BN2_BE),
        nel, 512);
  }

  // ======================= sa3 MLP + mean + head =======================
  concat3_kernel<<<cdiv(Bb * M4 * 516, 256), 256, 0, stream>>>(xtd2, posq4, f3, Bb * M4, 512);
  gemm(f3,  LW(IN_SA3, 0), LB(IN_SA3, 0), LG(IN_SA3, 0), LBE(IN_SA3, 0), 0, 0,
       h3a, Bb * M4, 256, 516, 515, 516, 1, 0);
  gemm(h3a, LW(IN_SA3, 1), LB(IN_SA3, 1), LG(IN_SA3, 1), LBE(IN_SA3, 1), 0, 0,
       h3b, Bb * M4, 512, 256, 256, 256, 1, 0);
  gemm(h3b, LW(IN_SA3, 2), LB(IN_SA3, 2), 0, 0, 0, 0,
       h3c, Bb * M4, 1024, 512, 512, 512, 0, 0);
  mean_kernel<<<cdiv(Bb * 1024, 256), 256, 0, stream>>>(h3c, pooled, Bb, M4, 1024);

  gemm(pooled, LW(IN_HEAD, 0), LB(IN_HEAD, 0), 0, 0, 0, 0, hh1, Bb, 512, 1024, 1024, 1024, 4, 0);
  gemm(hh1,    LW(IN_HEAD, 1), LB(IN_HEAD, 1), 0, 0, 0, 0, hh2, Bb, 256, 512, 512, 512, 4, 0);
  gemm(hh2,    LW(IN_HEAD, 2), LB(IN_HEAD, 2), 0, 0, 0, 0, (float*)d_out, Bb, 40, 256, 256, 256, 0, 0);
}